// GATLayer_58136677319341
// MI455X (gfx1250) — compile-verified
//
#include <hip/hip_runtime.h>
#include <cstdint>
#include <cstddef>

#define BETA   0.2f
#define GAMMA  0.95f
#define DFEAT  64
#define EPB    8                    // edges per LDS batch
#define BATCH_BYTES (EPB * 512)     // 8 edges * 128 floats * 4 B = 4 KB
#define WAVES_PER_BLOCK 8

typedef float v4f __attribute__((ext_vector_type(4)));

// ---------------------------------------------------------------------------
// Native f32 atomic add, no-return form, DEVICE scope (cross-WGP coherent).
// Inline asm guarantees global_atomic_add_f32 (never a CAS loop) and an
// explicit scope:SCOPE_DEV so the RMW resolves in the device-level L2.
// ---------------------------------------------------------------------------
__device__ __forceinline__ void atomAddF32(float* p, float v) {
  asm volatile("global_atomic_add_f32 %0, %1, off scope:SCOPE_DEV"
               :: "v"((unsigned long long)(uintptr_t)p), "v"(v)
               : "memory");
}

// ---------------------------------------------------------------------------
// CDNA5 async copy: stage one 4 KB edge batch (8 edges x 512 B) into LDS.
// Each lane moves 16 B per instruction; 8 instructions cover 4 KB.
// INST_OFFSET is added to BOTH the global and LDS address (ISA 08, §4.4).
// th:TH_LOAD_NT -> streamed-once edge data must not evict the L2-resident
// atomic accumulators.  Tracked by ASYNCcnt.
// ---------------------------------------------------------------------------
__device__ __forceinline__ void prefetch_batch(unsigned ldsA, unsigned long long gA) {
  asm volatile(
      "global_load_async_to_lds_b128 %0, %1, off th:TH_LOAD_NT\n\t"
      "global_load_async_to_lds_b128 %0, %1, off offset:512 th:TH_LOAD_NT\n\t"
      "global_load_async_to_lds_b128 %0, %1, off offset:1024 th:TH_LOAD_NT\n\t"
      "global_load_async_to_lds_b128 %0, %1, off offset:1536 th:TH_LOAD_NT\n\t"
      "global_load_async_to_lds_b128 %0, %1, off offset:2048 th:TH_LOAD_NT\n\t"
      "global_load_async_to_lds_b128 %0, %1, off offset:2560 th:TH_LOAD_NT\n\t"
      "global_load_async_to_lds_b128 %0, %1, off offset:3072 th:TH_LOAD_NT\n\t"
      "global_load_async_to_lds_b128 %0, %1, off offset:3584 th:TH_LOAD_NT"
      :: "v"(ldsA), "v"(gA) : "memory");
}

// Wave32 xor-butterfly max via ds_swizzle (group-of-32 mode: offset[14:10]=xor,
// [4:0]=and_mask=0x1f).  Single DS op per step, no bpermute index math.
__device__ __forceinline__ float waveMax32(float m) {
  m = fmaxf(m, __int_as_float(__builtin_amdgcn_ds_swizzle(__float_as_int(m), 0x401F))); // xor 16
  m = fmaxf(m, __int_as_float(__builtin_amdgcn_ds_swizzle(__float_as_int(m), 0x201F))); // xor 8
  m = fmaxf(m, __int_as_float(__builtin_amdgcn_ds_swizzle(__float_as_int(m), 0x101F))); // xor 4
  m = fmaxf(m, __int_as_float(__builtin_amdgcn_ds_swizzle(__float_as_int(m), 0x081F))); // xor 2
  m = fmaxf(m, __int_as_float(__builtin_amdgcn_ds_swizzle(__float_as_int(m), 0x041F))); // xor 1
  return m;
}

// ---------------------------------------------------------------------------
// Kernel 1: per-node max over D=64.  One wave32 per node, float2 per lane.
// ---------------------------------------------------------------------------
__global__ void __launch_bounds__(256)
zmax_kernel(const float* __restrict__ z, float* __restrict__ zmax, int n) {
  const int wave = threadIdx.x >> 5;
  const int lane = threadIdx.x & 31;
  const int node = blockIdx.x * WAVES_PER_BLOCK + wave;
  if (node >= n) return;
  const float2 v = ((const float2*)(z + (size_t)node * DFEAT))[lane];
  const float m = waveMax32(fmaxf(v.x, v.y));
  if (lane == 0) zmax[node] = m;
}

// ---------------------------------------------------------------------------
// Kernel 2: edge scatter.  One wave32 per 8-edge batch; double-buffered
// async-to-LDS pipeline; 4 native f32 atomics per lane per edge + 1 deg
// atomic per edge.  Accumulators (52 MB) are L2-resident on MI455X (192 MB).
// ---------------------------------------------------------------------------
__global__ void __launch_bounds__(256)
edge_kernel(const float* __restrict__ e,
            const int*   __restrict__ src,
            const int*   __restrict__ dst,
            const float* __restrict__ zmax,
            float* __restrict__ sum_q,
            float* __restrict__ sum_ac,
            float* __restrict__ deg,
            int nEdges) {
  __shared__ __align__(16) char smem[WAVES_PER_BLOCK][2 * BATCH_BYTES]; // 64 KB

  const int wave   = threadIdx.x >> 5;
  const int lane   = threadIdx.x & 31;
  const int gwave  = blockIdx.x * WAVES_PER_BLOCK + wave;
  const int nWaves = gridDim.x * WAVES_PER_BLOCK;
  const int nBatches = nEdges / EPB;

  // Low 32 bits of an AS(3) pointer == LDS byte offset used by DS/async ops.
  const unsigned ldsBase =
      (unsigned)(uintptr_t)(&smem[wave][0]) + (unsigned)(lane * 16);
  const unsigned long long gBase =
      (unsigned long long)(uintptr_t)e + (unsigned long long)(lane * 16);

  int b = gwave;
  if (b < nBatches)
    prefetch_batch(ldsBase, gBase + (unsigned long long)b * BATCH_BYTES);

  int it = 0;
  for (; b < nBatches; b += nWaves, ++it) {
    const int nb = b + nWaves;
    const bool pf = (nb < nBatches);
    if (pf)
      prefetch_batch(ldsBase + ((it + 1) & 1) * BATCH_BYTES,
                     gBase + (unsigned long long)nb * BATCH_BYTES);
    // In-order async completion: <=8 outstanding means current batch landed.
    if (pf) asm volatile("s_wait_asynccnt 0x8" ::: "memory");
    else    asm volatile("s_wait_asynccnt 0x0" ::: "memory");

    const float* buf = (const float*)(smem[wave] + (it & 1) * BATCH_BYTES);
#pragma unroll
    for (int k = 0; k < EPB; ++k) {
      const int eIdx = b * EPB + k;
      const int s = src[eIdx];                 // broadcast load
      const int d = dst[eIdx];
      const float zm = GAMMA * zmax[s];
      const float2 e0 = *(const float2*)(buf + k * 128 + lane * 2);
      const float2 e1 = *(const float2*)(buf + k * 128 + 64 + lane * 2);
      const float qx = fmaf(zm, e1.x, e0.x);
      const float qy = fmaf(zm, e1.y, e0.y);
      float* sq = sum_q  + (size_t)(unsigned)d * DFEAT + lane * 2;
      float* sa = sum_ac + (size_t)(unsigned)d * DFEAT + lane * 2;
      atomAddF32(sq,     qx);
      atomAddF32(sq + 1, qy);
      atomAddF32(sa,     e1.x);
      atomAddF32(sa + 1, e1.y);
      if (lane == 0) atomAddF32(deg + d, 1.0f);
    }
  }

  // Tail edges (nEdges % 8), straight from global (none for N_EDGES=1.6M).
  for (int eIdx = nBatches * EPB + gwave; eIdx < nEdges; eIdx += nWaves) {
    const int s = src[eIdx];
    const int d = dst[eIdx];
    const float zm = GAMMA * zmax[s];
    const float2 e0 = *(const float2*)(e + (size_t)eIdx * 128 + lane * 2);
    const float2 e1 = *(const float2*)(e + (size_t)eIdx * 128 + 64 + lane * 2);
    const float qx = fmaf(zm, e1.x, e0.x);
    const float qy = fmaf(zm, e1.y, e0.y);
    float* sq = sum_q  + (size_t)(unsigned)d * DFEAT + lane * 2;
    float* sa = sum_ac + (size_t)(unsigned)d * DFEAT + lane * 2;
    atomAddF32(sq,     qx);
    atomAddF32(sq + 1, qy);
    atomAddF32(sa,     e1.x);
    atomAddF32(sa + 1, e1.y);
    if (lane == 0) atomAddF32(deg + d, 1.0f);
  }
}

// ---------------------------------------------------------------------------
// Kernel 3: z_new = beta*z + (1-beta)*sum_q/(sum_ac+1e-6), gated by deg>0.
// float4 per thread.  sum_q/sum_ac are dead after this read -> NT loads;
// out is never re-read on device -> NT store.
// ---------------------------------------------------------------------------
__global__ void __launch_bounds__(256)
finalize_kernel(const float* __restrict__ z,
                const float* __restrict__ sum_q,
                const float* __restrict__ sum_ac,
                const float* __restrict__ deg,
                float* __restrict__ out, int n) {
  const int i = blockIdx.x * blockDim.x + threadIdx.x;  // one float4
  if (i >= n * (DFEAT / 4)) return;
  const int node = i >> 4;
  const v4f z4 = ((const v4f*)z)[i];
  v4f r = z4;
  if (deg[node] > 0.0f) {
    const v4f q4 = __builtin_nontemporal_load(((const v4f*)sum_q) + i);
    const v4f a4 = __builtin_nontemporal_load(((const v4f*)sum_ac) + i);
    r.x = BETA * z4.x + (1.0f - BETA) * q4.x / (a4.x + 1e-6f);
    r.y = BETA * z4.y + (1.0f - BETA) * q4.y / (a4.y + 1e-6f);
    r.z = BETA * z4.z + (1.0f - BETA) * q4.z / (a4.z + 1e-6f);
    r.w = BETA * z4.w + (1.0f - BETA) * q4.w / (a4.w + 1e-6f);
  }
  __builtin_nontemporal_store(r, ((v4f*)out) + i);
}

// ---------------------------------------------------------------------------
extern "C" void kernel_launch(void* const* d_in, const int* in_sizes, int n_in,
                              void* d_out, int out_size, void* d_ws, size_t ws_size,
                              hipStream_t stream) {
  const float* z   = (const float*)d_in[0];
  const float* e   = (const float*)d_in[1];
  const int*   src = (const int*)d_in[2];
  const int*   dst = (const int*)d_in[3];
  float*       out = (float*)d_out;

  const int n      = in_sizes[0] / DFEAT;   // 100000
  const int nEdges = in_sizes[2];           // 1600000

  // Workspace layout: sum_q[n*64] | sum_ac[n*64] | deg[n] | zmax[n]
  float* sum_q  = (float*)d_ws;
  float* sum_ac = sum_q  + (size_t)n * DFEAT;
  float* degp   = sum_ac + (size_t)n * DFEAT;
  float* zmaxp  = degp + n;

  // Zero the accumulators (graph-capturable). zmax is fully overwritten.
  hipMemsetAsync(d_ws, 0, ((size_t)n * 2 * DFEAT + n) * sizeof(float), stream);

  zmax_kernel<<<(n + WAVES_PER_BLOCK - 1) / WAVES_PER_BLOCK, 256, 0, stream>>>(
      z, zmaxp, n);

  // 2048 blocks * 8 waves = 16384 waves -> ~12 batches per wave.
  edge_kernel<<<2048, 256, 0, stream>>>(e, src, dst, zmaxp,
                                        sum_q, sum_ac, degp, nEdges);

  const int nVec = n * (DFEAT / 4);
  finalize_kernel<<<(nVec + 255) / 256, 256, 0, stream>>>(
      z, sum_q, sum_ac, degp, out, n);
}